// DeepGATBlockV1_26637387170227
// MI455X (gfx1250) — compile-verified
//
#include <hip/hip_runtime.h>

#define N_NODES 50000
#define N_EDGES 400000
#define C_DIM   64
#define H_HEADS 4
#define HC      256          // H*C
#define ET      (N_EDGES + N_NODES)   // edges + self loops

typedef __attribute__((ext_vector_type(2))) float v2f;
typedef __attribute__((ext_vector_type(8))) float v8f;

__device__ __forceinline__ float waveReduceSum(float v) {
  #pragma unroll
  for (int off = 16; off > 0; off >>= 1)
    v += __shfl_xor(v, off, 32);
  return v;
}

// IEEE bit-trick float atomic max (works for mixed signs).
__device__ __forceinline__ void atomicMaxF(float* addr, float val) {
  if (val >= 0.0f) atomicMax((int*)addr, __float_as_int(val));
  else             atomicMin((unsigned int*)addr, (unsigned int)__float_as_int(val));
}

// ---------------------------------------------------------------------------
// Convert int64 edge list -> int32, append self loops.
// ---------------------------------------------------------------------------
__global__ __launch_bounds__(256) void prep_edges(const long long* __restrict__ ei,
                                                  int* __restrict__ src32,
                                                  int* __restrict__ dst32) {
  int t = blockIdx.x * blockDim.x + threadIdx.x;
  if (t < N_EDGES) {
    src32[t] = (int)ei[t];
    dst32[t] = (int)ei[N_EDGES + t];
  } else if (t < ET) {
    src32[t] = t - N_EDGES;
    dst32[t] = t - N_EDGES;
  }
}

// ---------------------------------------------------------------------------
// C[M,Nt] = A[M,64] @ B[64,Nt] (+bias) (+relu) using V_WMMA_F32_16X16X4_F32.
// One block = 8 waves = one 16-row M tile; waves stripe over Nt/16 tiles.
// B fragments for the whole K range are preloaded into VGPRs so the inner
// loop is 16 back-to-back WMMAs on registers (one load-wait per tile).
// ---------------------------------------------------------------------------
__global__ __launch_bounds__(256) void gemm_k64_wmma(
    const float* __restrict__ A, const float* __restrict__ B,
    const float* __restrict__ bias, float* __restrict__ Cout,
    int M, int Nt, int relu) {
  __shared__ float As[16 * 65];  // +1 pad: conflict-free per-row access
  const int row0 = blockIdx.x * 16;

  for (int idx = threadIdx.x; idx < 16 * 64; idx += 256) {
    int r = idx >> 6, c = idx & 63;
    int gr = row0 + r;
    As[r * 65 + c] = (gr < M) ? A[gr * 64 + c] : 0.0f;
  }
  __syncthreads();

  const int wave = threadIdx.x >> 5;
  const int lane = threadIdx.x & 31;
  const int mrow = lane & 15;
  const int koff = (lane < 16) ? 0 : 2;  // ISA A-layout: lanes 0-15 K0/K1, 16-31 K2/K3
  const int ncol = lane & 15;
  const int ntiles = Nt >> 4;

  // A fragments (loop-invariant across jt): hoist once into registers.
  float afrag[32];
  #pragma unroll
  for (int k = 0; k < 16; ++k) {
    afrag[2 * k]     = As[mrow * 65 + k * 4 + koff];
    afrag[2 * k + 1] = As[mrow * 65 + k * 4 + koff + 1];
  }

  for (int jt = wave; jt < ntiles; jt += 8) {
    // Preload all B fragments: 32 loads in flight, single wait.
    float bfrag[32];
    #pragma unroll
    for (int k = 0; k < 16; ++k) {
      const int kb = k * 4 + koff;
      bfrag[2 * k]     = B[kb * Nt + jt * 16 + ncol];
      bfrag[2 * k + 1] = B[(kb + 1) * Nt + jt * 16 + ncol];
    }
    v8f acc = {};
    #pragma unroll
    for (int k = 0; k < 16; ++k) {
      v2f a, b;
      a[0] = afrag[2 * k];
      a[1] = afrag[2 * k + 1];
      b[0] = bfrag[2 * k];
      b[1] = bfrag[2 * k + 1];
      acc = __builtin_amdgcn_wmma_f32_16x16x4_f32(
          false, a, false, b, (short)0, acc, false, false);
    }
    const int rbase = row0 + ((lane < 16) ? 0 : 8);
    const int col   = jt * 16 + ncol;
    const float bv  = bias ? bias[col] : 0.0f;
    #pragma unroll
    for (int v = 0; v < 8; ++v) {
      int r = rbase + v;
      if (r < M) {
        float val = acc[v] + bv;
        if (relu) val = fmaxf(val, 0.0f);
        Cout[r * Nt + col] = val;
      }
    }
  }
}

// ---------------------------------------------------------------------------
// s_src/s_dst: wave per node; lane handles 8 consecutive channels (float4 x2),
// segmented 8-lane shuffle reduction per head. Fully coalesced.
// ---------------------------------------------------------------------------
__global__ __launch_bounds__(256) void gat_scores(
    const float* __restrict__ xh, const float* __restrict__ a_src,
    const float* __restrict__ a_dst, float* __restrict__ ssrc,
    float* __restrict__ sdst) {
  int lane = threadIdx.x & 31;
  int n = blockIdx.x * 8 + (threadIdx.x >> 5);
  if (n >= N_NODES) return;
  int h = lane >> 3;  // lanes 0-7 -> head 0, ... 24-31 -> head 3
  const float4* xr = (const float4*)(xh + (size_t)n * HC + lane * 8);
  float4 x0 = xr[0], x1 = xr[1];
  const float4* s4 = (const float4*)(a_src + h * 64 + (lane & 7) * 8);
  const float4* d4 = (const float4*)(a_dst + h * 64 + (lane & 7) * 8);
  float4 as0 = s4[0], as1 = s4[1];
  float4 ad0 = d4[0], ad1 = d4[1];
  float s1 = x0.x * as0.x + x0.y * as0.y + x0.z * as0.z + x0.w * as0.w
           + x1.x * as1.x + x1.y * as1.y + x1.z * as1.z + x1.w * as1.w;
  float s2 = x0.x * ad0.x + x0.y * ad0.y + x0.z * ad0.z + x0.w * ad0.w
           + x1.x * ad1.x + x1.y * ad1.y + x1.z * ad1.z + x1.w * ad1.w;
  #pragma unroll
  for (int m = 1; m <= 4; m <<= 1) {
    s1 += __shfl_xor(s1, m, 32);
    s2 += __shfl_xor(s2, m, 32);
  }
  if ((lane & 7) == 0) {
    ssrc[n * 4 + h] = s1;
    sdst[n * 4 + h] = s2;
  }
}

__global__ __launch_bounds__(256) void gat_init(float* __restrict__ m,
                                                float* __restrict__ z,
                                                float* __restrict__ agg) {
  int t = blockIdx.x * blockDim.x + threadIdx.x;
  if (t < N_NODES * H_HEADS) { m[t] = -3.0e38f; z[t] = 0.0f; }
  if (t < N_NODES * HC) agg[t] = 0.0f;
}

// pass 1: e = leaky_relu(s_src[src]+s_dst[dst]); m[dst] = max
__global__ __launch_bounds__(256) void edge_scores_max(
    const int* __restrict__ src, const int* __restrict__ dst,
    const float* __restrict__ ssrc, const float* __restrict__ sdst,
    float* __restrict__ ebuf, float* __restrict__ m) {
  int t = blockIdx.x * blockDim.x + threadIdx.x;   // t = e*4 + h
  if (t >= ET * H_HEADS) return;
  int e = t >> 2, h = t & 3;
  int s = src[e], d = dst[e];
  float v = ssrc[s * 4 + h] + sdst[d * 4 + h];
  v = (v > 0.0f) ? v : 0.2f * v;
  ebuf[t] = v;
  atomicMaxF(&m[d * 4 + h], v);
}

// pass 2: ee = exp(e - m[dst]); z[dst] += ee
__global__ __launch_bounds__(256) void edge_exp_sum(
    const int* __restrict__ dst, float* __restrict__ ebuf,
    const float* __restrict__ m, float* __restrict__ z) {
  int t = blockIdx.x * blockDim.x + threadIdx.x;
  if (t >= ET * H_HEADS) return;
  int e = t >> 2, h = t & 3;
  int d = dst[e];
  float ee = __expf(ebuf[t] - m[d * 4 + h]);
  ebuf[t] = ee;
  atomicAdd(&z[d * 4 + h], ee);
}

// pass 3: agg[dst] += alpha * xh[src]; one wave per edge, 8 chans/lane.
__global__ __launch_bounds__(256) void edge_scatter(
    const int* __restrict__ src, const int* __restrict__ dst,
    const float* __restrict__ ebuf, const float* __restrict__ z,
    const float* __restrict__ xh, float* __restrict__ agg) {
  int lane = threadIdx.x & 31;
  int e = blockIdx.x * 8 + (threadIdx.x >> 5);
  if (e >= ET) return;
  int s = src[e], d = dst[e];
  int h = lane >> 3;                               // (lane*8)/64
  float alpha = ebuf[e * 4 + h] / (z[d * 4 + h] + 1e-16f);
  const float4* xs = (const float4*)(xh + (size_t)s * HC + lane * 8);
  float4 v0 = xs[0], v1 = xs[1];
  float* ag = agg + (size_t)d * HC + lane * 8;
  atomicAdd(ag + 0, v0.x * alpha);
  atomicAdd(ag + 1, v0.y * alpha);
  atomicAdd(ag + 2, v0.z * alpha);
  atomicAdd(ag + 3, v0.w * alpha);
  atomicAdd(ag + 4, v1.x * alpha);
  atomicAdd(ag + 5, v1.y * alpha);
  atomicAdd(ag + 6, v1.z * alpha);
  atomicAdd(ag + 7, v1.w * alpha);
}

// head-mean + att_bias + residual + LayerNorm. One wave per node.
__global__ __launch_bounds__(256) void gat_finish_ln(
    const float* __restrict__ xin, const float* __restrict__ agg,
    const float* __restrict__ att_bias, const float* __restrict__ g,
    const float* __restrict__ b, float* __restrict__ out) {
  int lane = threadIdx.x & 31;
  int n = blockIdx.x * 8 + (threadIdx.x >> 5);
  if (n >= N_NODES) return;
  float pre[2];
  #pragma unroll
  for (int j = 0; j < 2; ++j) {
    int c = lane + j * 32;
    const float* ar = agg + (size_t)n * HC + c;
    float av = (ar[0] + ar[64] + ar[128] + ar[192]) * 0.25f + att_bias[c];
    pre[j] = xin[n * 64 + c] + av;
  }
  float mu = waveReduceSum(pre[0] + pre[1]) * (1.0f / 64.0f);
  float d0 = pre[0] - mu, d1 = pre[1] - mu;
  float var = waveReduceSum(d0 * d0 + d1 * d1) * (1.0f / 64.0f);
  float rstd = rsqrtf(var + 1e-5f);
  out[n * 64 + lane]      = d0 * rstd * g[lane]      + b[lane];
  out[n * 64 + lane + 32] = d1 * rstd * g[lane + 32] + b[lane + 32];
}

// residual + LayerNorm. One wave per node.
__global__ __launch_bounds__(256) void add_ln(
    const float* __restrict__ xres, const float* __restrict__ hadd,
    const float* __restrict__ g, const float* __restrict__ b,
    float* __restrict__ out) {
  int lane = threadIdx.x & 31;
  int n = blockIdx.x * 8 + (threadIdx.x >> 5);
  if (n >= N_NODES) return;
  float p0 = xres[n * 64 + lane]      + hadd[n * 64 + lane];
  float p1 = xres[n * 64 + lane + 32] + hadd[n * 64 + lane + 32];
  float mu = waveReduceSum(p0 + p1) * (1.0f / 64.0f);
  float d0 = p0 - mu, d1 = p1 - mu;
  float var = waveReduceSum(d0 * d0 + d1 * d1) * (1.0f / 64.0f);
  float rstd = rsqrtf(var + 1e-5f);
  out[n * 64 + lane]      = d0 * rstd * g[lane]      + b[lane];
  out[n * 64 + lane + 32] = d1 * rstd * g[lane + 32] + b[lane + 32];
}

// ---------------------------------------------------------------------------
extern "C" void kernel_launch(void* const* d_in, const int* in_sizes, int n_in,
                              void* d_out, int out_size, void* d_ws, size_t ws_size,
                              hipStream_t stream) {
  (void)in_sizes; (void)n_in; (void)out_size; (void)ws_size;

  const float*     x        = (const float*)d_in[0];
  const long long* ei       = (const long long*)d_in[1];
  const float*     Wg       = (const float*)d_in[2];
  const float*     att_src  = (const float*)d_in[3];
  const float*     att_dst  = (const float*)d_in[4];
  const float*     att_bias = (const float*)d_in[5];
  const float*     ln1_g    = (const float*)d_in[6];
  const float*     ln1_b    = (const float*)d_in[7];
  const float*     W1       = (const float*)d_in[8];
  const float*     b1       = (const float*)d_in[9];
  const float*     W2       = (const float*)d_in[10];
  const float*     b2       = (const float*)d_in[11];
  const float*     ln2_g    = (const float*)d_in[12];
  const float*     ln2_b    = (const float*)d_in[13];
  float* out = (float*)d_out;

  char* ws = (char*)d_ws;
  size_t off = 0;
  auto alloc = [&](size_t bytes) -> void* {
    void* p = (void*)(ws + off);
    off += (bytes + 255) & ~size_t(255);
    return p;
  };

  int*   src32 = (int*)alloc((size_t)ET * sizeof(int));
  int*   dst32 = (int*)alloc((size_t)ET * sizeof(int));
  float* xcur  = (float*)alloc((size_t)N_NODES * C_DIM * sizeof(float));
  float* x1    = (float*)alloc((size_t)N_NODES * C_DIM * sizeof(float));
  float* xh    = (float*)alloc((size_t)N_NODES * HC * sizeof(float));
  float* agg   = (float*)alloc((size_t)N_NODES * HC * sizeof(float));
  float* ssrc  = (float*)alloc((size_t)N_NODES * H_HEADS * sizeof(float));
  float* sdst  = (float*)alloc((size_t)N_NODES * H_HEADS * sizeof(float));
  float* mbuf  = (float*)alloc((size_t)N_NODES * H_HEADS * sizeof(float));
  float* zbuf  = (float*)alloc((size_t)N_NODES * H_HEADS * sizeof(float));
  float* ebuf  = (float*)alloc((size_t)ET * H_HEADS * sizeof(float));
  float* hbuf  = xh;   // FFN hidden aliases xh (dead by then)
  float* h2    = agg;  // FFN output aliases agg (dead by then)

  const int mtiles = (N_NODES + 15) / 16;  // 3125 exactly

  prep_edges<<<(ET + 255) / 256, 256, 0, stream>>>(ei, src32, dst32);
  hipMemcpyAsync(xcur, x, (size_t)N_NODES * C_DIM * sizeof(float),
                 hipMemcpyDeviceToDevice, stream);

  for (int i = 0; i < 2; ++i) {
    const float* Wg_i  = Wg + (size_t)i * C_DIM * HC;
    const float* as_i  = att_src + (size_t)i * H_HEADS * C_DIM;
    const float* ad_i  = att_dst + (size_t)i * H_HEADS * C_DIM;
    const float* ab_i  = att_bias + (size_t)i * C_DIM;
    const float* l1g_i = ln1_g + (size_t)i * C_DIM;
    const float* l1b_i = ln1_b + (size_t)i * C_DIM;
    const float* W1_i  = W1 + (size_t)i * C_DIM * C_DIM;
    const float* b1_i  = b1 + (size_t)i * C_DIM;
    const float* W2_i  = W2 + (size_t)i * C_DIM * C_DIM;
    const float* b2_i  = b2 + (size_t)i * C_DIM;
    const float* l2g_i = ln2_g + (size_t)i * C_DIM;
    const float* l2b_i = ln2_b + (size_t)i * C_DIM;

    // xh = xcur @ Wg[i]   [N,256]
    gemm_k64_wmma<<<mtiles, 256, 0, stream>>>(xcur, Wg_i, nullptr, xh,
                                              N_NODES, HC, 0);
    gat_scores<<<(N_NODES + 7) / 8, 256, 0, stream>>>(xh, as_i, ad_i, ssrc,
                                                      sdst);
    gat_init<<<(N_NODES * HC + 255) / 256, 256, 0, stream>>>(mbuf, zbuf, agg);
    edge_scores_max<<<(ET * H_HEADS + 255) / 256, 256, 0, stream>>>(
        src32, dst32, ssrc, sdst, ebuf, mbuf);
    edge_exp_sum<<<(ET * H_HEADS + 255) / 256, 256, 0, stream>>>(
        dst32, ebuf, mbuf, zbuf);
    edge_scatter<<<(ET + 7) / 8, 256, 0, stream>>>(src32, dst32, ebuf, zbuf,
                                                   xh, agg);
    // x1 = LN(xcur + mean_heads(agg) + att_bias)
    gat_finish_ln<<<(N_NODES + 7) / 8, 256, 0, stream>>>(xcur, agg, ab_i,
                                                         l1g_i, l1b_i, x1);
    // FFN
    gemm_k64_wmma<<<mtiles, 256, 0, stream>>>(x1, W1_i, b1_i, hbuf,
                                              N_NODES, C_DIM, 1);
    gemm_k64_wmma<<<mtiles, 256, 0, stream>>>(hbuf, W2_i, b2_i, h2,
                                              N_NODES, C_DIM, 0);
    float* dst_x = (i == 1) ? out : xcur;
    add_ln<<<(N_NODES + 7) / 8, 256, 0, stream>>>(x1, h2, l2g_i, l2b_i, dst_x);
  }
}